// set2set_aggr_80264348827990
// MI455X (gfx1250) — compile-verified
//
#include <hip/hip_runtime.h>
#include <hip/hip_bf16.h>
#include <math.h>

// Set2Set aggregation for MI455X (gfx1250, wave32).
//  - LSTM gate GEMMs on the matrix pipe: v_wmma_f32_16x16x4_f32, 4 M-tiles per
//    wave so each weight (B) fragment feeds 4 WMMAs.
//  - Attention is the roofline term: x is 256MB (doesn't fit 192MB L2). Pass A
//    streams x once per step and spills segment rows to LDS; pass C re-reads
//    from LDS (global fallback only for >CAP-row segments). ~768MB total HBM
//    traffic -> ~33us at 23.3 TB/s.

typedef float v2f __attribute__((ext_vector_type(2)));
typedef float v8f __attribute__((ext_vector_type(8)));

#define PROCESSING_STEPS 3
#define SEG_CAP 150   // LDS-cached rows per segment: 150*512B = 75KB -> 4 WGs/WGP

// ---------------------------------------------------------------------------
// GEMM: gates[G,4D] = qstar[G,2D] @ W_ih^T + h[G,D] @ W_hh^T + (b_ih + b_hh)
// One wave computes a 64x16 strip (4 stacked 16x16 tiles) so the B fragment is
// reused across 4 WMMAs per K-step.
// A 16x4 f32 layout: lanes 0-15 row M=l, VGPR0=K0,VGPR1=K1; lanes 16-31 K2/K3.
// B 4x16 f32 layout mirrors it; C/D: VGPR r, lane half selects M=r / M=r+8.
// ---------------------------------------------------------------------------
__device__ __forceinline__ void wmma_panel4(const float* __restrict__ A, int lda,
                                            const float* __restrict__ W, int ldw,
                                            int row0, int col0, int K,
                                            v8f acc[4], int l, int half) {
  const float* wrow = W + (size_t)(col0 + l) * ldw + 2 * half;
  for (int kk = 0; kk < K; kk += 4) {
    v2f b;
    b[0] = wrow[kk];
    b[1] = wrow[kk + 1];
#pragma unroll
    for (int t = 0; t < 4; ++t) {
      const float* arow = A + (size_t)(row0 + t * 16 + l) * lda + 2 * half;
      v2f a;
      a[0] = arow[kk];
      a[1] = arow[kk + 1];
      acc[t] = __builtin_amdgcn_wmma_f32_16x16x4_f32(false, a, false, b,
                                                     (short)0, acc[t], false, false);
    }
  }
}

__global__ void gemm_gates_kernel(const float* __restrict__ qstar,
                                  const float* __restrict__ h,
                                  const float* __restrict__ W_ih,
                                  const float* __restrict__ W_hh,
                                  const float* __restrict__ b_ih,
                                  const float* __restrict__ b_hh,
                                  float* __restrict__ gates,
                                  int G, int D) {
  const int lane = threadIdx.x;          // 0..31
  const int l    = lane & 15;
  const int half = lane >> 4;
  const int row0 = blockIdx.x * 64;                       // over G (4 tiles)
  const int col0 = (blockIdx.y * 4 + threadIdx.y) * 16;   // over 4D
  const int twoD = 2 * D, fourD = 4 * D;

  const float bias = b_ih[col0 + l] + b_hh[col0 + l];
  v8f acc[4];
#pragma unroll
  for (int t = 0; t < 4; ++t)
#pragma unroll
    for (int r = 0; r < 8; ++r) acc[t][r] = bias;

  wmma_panel4(qstar, twoD, W_ih, twoD, row0, col0, twoD, acc, l, half);
  wmma_panel4(h,     D,    W_hh, D,    row0, col0, D,    acc, l, half);

#pragma unroll
  for (int t = 0; t < 4; ++t)
#pragma unroll
    for (int r = 0; r < 8; ++r) {
      const int row = row0 + t * 16 + r + half * 8;
      gates[(size_t)row * fourD + col0 + l] = acc[t][r];
    }
}

// ---------------------------------------------------------------------------
// Pointwise LSTM cell update; also writes q_star[:, :D] = h.
// ---------------------------------------------------------------------------
__device__ __forceinline__ float sigf(float v) { return 1.0f / (1.0f + expf(-v)); }

__global__ void lstm_pointwise_kernel(const float* __restrict__ gates,
                                      float* __restrict__ c,
                                      float* __restrict__ h,
                                      float* __restrict__ qstar,
                                      int G, int D) {
  const int idx = blockIdx.x * blockDim.x + threadIdx.x;
  if (idx >= G * D) return;
  const int g = idx / D;
  const int d = idx - g * D;
  const float* gr = gates + (size_t)g * 4 * D;
  const float gi = gr[d];
  const float gf = gr[D + d];
  const float gg = gr[2 * D + d];
  const float go = gr[3 * D + d];
  const float cn = sigf(gf) * c[idx] + sigf(gi) * tanhf(gg);
  c[idx] = cn;
  const float hn = sigf(go) * tanhf(cn);
  h[idx] = hn;
  qstar[(size_t)g * (2 * D) + d] = hn;   // q half of q_star
}

// ---------------------------------------------------------------------------
// Per-segment attention: one block (256 threads = 8 wave32) per segment g.
// Dynamic LDS caches up to SEG_CAP rows of x during pass A so pass C does not
// re-stream HBM. e values staged in a small global scratch (2MB, L2-resident).
// ---------------------------------------------------------------------------
__global__ void attn_segment_kernel(const float* __restrict__ x,
                                    const int* __restrict__ index,
                                    float* __restrict__ qstar,
                                    float* __restrict__ e_buf,
                                    int N, int D) {
  extern __shared__ float smem[];
  float* xs   = smem;                         // [SEG_CAP * 128]
  float* q_s  = xs + SEG_CAP * 128;           // [128]
  float* r_s  = q_s + 128;                    // [8 * 128]
  float* red  = r_s + 8 * 128;                // [8]
  float* scal = red + 8;                      // [2]
  int*   seg  = (int*)(scal + 2);             // [2]

  const int g   = blockIdx.x;
  const int tid = threadIdx.x;

  if (tid == 0) {
    int lo = 0, hi = N;                       // lower_bound(index, g)
    while (lo < hi) { int mid = (lo + hi) >> 1; if (index[mid] < g) lo = mid + 1; else hi = mid; }
    seg[0] = lo;
    hi = N;                                   // lower_bound(index, g+1)
    while (lo < hi) { int mid = (lo + hi) >> 1; if (index[mid] < g + 1) lo = mid + 1; else hi = mid; }
    seg[1] = lo;
  }
  __syncthreads();
  const int start = seg[0], end = seg[1];

  if (tid < 128) q_s[tid] = qstar[(size_t)g * 256 + tid];  // q_g = h_g
  __syncthreads();

  const int wave = tid >> 5;
  const int lane = tid & 31;

  // ---- Pass A: stream x once; dot products + LDS spill + running max
  float lmax = -INFINITY;
  for (int row = start + wave; row < end; row += 8) {
    if (row + 8 < end)  // prefetch next row for this wave (global_prefetch)
      __builtin_prefetch(x + (size_t)(row + 8) * 128 + lane * 4, 0, 1);
    const float4 v = ((const float4*)(x + (size_t)row * 128))[lane];
    const int li = row - start;
    if (li < SEG_CAP) ((float4*)xs)[li * 32 + lane] = v;   // cache row in LDS
    float d = v.x * q_s[lane * 4 + 0] + v.y * q_s[lane * 4 + 1] +
              v.z * q_s[lane * 4 + 2] + v.w * q_s[lane * 4 + 3];
#pragma unroll
    for (int off = 16; off >= 1; off >>= 1) d += __shfl_xor(d, off, 32);
    if (lane == 0) { e_buf[row] = d; lmax = fmaxf(lmax, d); }
  }
  if (lane == 0) red[wave] = lmax;
  __syncthreads();
  if (tid == 0) {
    float m = red[0];
#pragma unroll
    for (int w = 1; w < 8; ++w) m = fmaxf(m, red[w]);
    if (!(m > -INFINITY)) m = 0.0f;          // empty-segment guard (matches ref)
    scal[0] = m;
  }
  __syncthreads();
  const float m = scal[0];

  // ---- Pass B: exp + sum over e (small; L2-resident)
  float lsum = 0.0f;
  for (int r2 = start + tid; r2 < end; r2 += 256) {
    const float ex = expf(e_buf[r2] - m);
    e_buf[r2] = ex;
    lsum += ex;
  }
#pragma unroll
  for (int off = 16; off >= 1; off >>= 1) lsum += __shfl_xor(lsum, off, 32);
  if (lane == 0) red[wave] = lsum;
  __syncthreads();
  if (tid == 0) {
    float s = 0.0f;
#pragma unroll
    for (int w = 0; w < 8; ++w) s += red[w];
    scal[1] = 1.0f / fmaxf(s, 1e-16f);
  }
  __syncthreads();
  const float inv_s = scal[1];

  // ---- Pass C: weighted sum of rows, served from LDS (global only if >CAP)
  float4 acc = make_float4(0.f, 0.f, 0.f, 0.f);
  for (int row = start + wave; row < end; row += 8) {
    const int li = row - start;
    const float4 v = (li < SEG_CAP)
        ? ((const float4*)xs)[li * 32 + lane]
        : ((const float4*)(x + (size_t)row * 128))[lane];
    const float a = e_buf[row] * inv_s;
    acc.x += v.x * a; acc.y += v.y * a; acc.z += v.z * a; acc.w += v.w * a;
  }
  r_s[wave * 128 + lane * 4 + 0] = acc.x;
  r_s[wave * 128 + lane * 4 + 1] = acc.y;
  r_s[wave * 128 + lane * 4 + 2] = acc.z;
  r_s[wave * 128 + lane * 4 + 3] = acc.w;
  __syncthreads();
  if (tid < 128) {
    float r = 0.0f;
#pragma unroll
    for (int w = 0; w < 8; ++w) r += r_s[w * 128 + tid];
    qstar[(size_t)g * 256 + 128 + tid] = r;  // r half of q_star
  }
}

// ---------------------------------------------------------------------------
extern "C" void kernel_launch(void* const* d_in, const int* in_sizes, int n_in,
                              void* d_out, int out_size, void* d_ws, size_t ws_size,
                              hipStream_t stream) {
  const float* x     = (const float*)d_in[0];
  const float* W_ih  = (const float*)d_in[1];
  const float* W_hh  = (const float*)d_in[2];
  const float* b_ih  = (const float*)d_in[3];
  const float* b_hh  = (const float*)d_in[4];
  const int*   index = (const int*)d_in[5];
  // d_in[6] = dim_size scalar; derived from out_size instead (host-known).

  const int D = 128;
  const int N = in_sizes[0] / D;          // 500000
  const int G = out_size / (2 * D);       // 4096

  float* qstar = (float*)d_out;                      // [G, 2D] -- doubles as output
  float* h     = (float*)d_ws;                       // [G, D]
  float* c     = h + (size_t)G * D;                  // [G, D]
  float* gates = c + (size_t)G * D;                  // [G, 4D]
  float* e_buf = gates + (size_t)G * 4 * D;          // [N]

  // zero-init recurrent state + q_star (graph-capture-safe async memset)
  hipMemsetAsync(qstar, 0, (size_t)out_size * sizeof(float), stream);
  hipMemsetAsync(h, 0, (size_t)G * D * 2 * sizeof(float), stream);  // h and c

  const dim3 gemm_grid(G / 64, (4 * D) / 64);
  const dim3 gemm_block(32, 4);
  const int total = G * D;
  const size_t attn_smem =
      (size_t)(SEG_CAP * 128 + 128 + 8 * 128 + 8 + 2 + 2) * sizeof(float);

  for (int step = 0; step < PROCESSING_STEPS; ++step) {
    gemm_gates_kernel<<<gemm_grid, gemm_block, 0, stream>>>(
        qstar, h, W_ih, W_hh, b_ih, b_hh, gates, G, D);
    lstm_pointwise_kernel<<<(total + 255) / 256, 256, 0, stream>>>(
        gates, c, h, qstar, G, D);
    attn_segment_kernel<<<G, 256, attn_smem, stream>>>(x, index, qstar, e_buf, N, D);
  }
}